// MyGraphPooling_70686571758193
// MI455X (gfx1250) — compile-verified
//
#include <hip/hip_runtime.h>
#include <hip/hip_bf16.h>

// Graph voxel pooling for MI455X (gfx1250, wave32).
//
// Pipeline:
//  1) voxel keys (24-bit key space!) -> dense presence bitmap P[16M]
//  2) exclusive scan of P -> ranks R (== jnp.unique sorted ranks), U = total
//  3) decode unique voxels / new_batch from (P,R)
//  4) inv[n] = R[key[n]]
//  5) pooled = segment_max via int-trick atomics; x tile staged through LDS
//     with the Tensor Data Mover (tensor_load_to_lds + s_wait_tensorcnt) when
//     the builtin exists, else per-lane global_load_async_to_lds_b128.
//  6) edge keys a*1e6+b (BIG=1e12 for self loops), 40-bit LSD radix sort
//     (4-bit digits x 10 passes), head-flag dedup + scan + scatter, self loops
//
// Output layout in d_out (tuple flattened in return order):
//   float32 pooled_x[N*F] | int64 uniq_qpos[N*3] | int64 ei_out[(E+N)*2] | int64 new_batch[N]

typedef unsigned long long u64;
typedef unsigned int u32;

#define NN 1000000
#define FF 64
#define EE 8000000
#define BBATCH 16
#define KXC 128
#define KYC 128
#define KTC 64
#define KEYSPACE (BBATCH * KXC * KYC * KTC)   // 16,777,216 = 2^24
#define BIGKEY 1000000000000ull               // N*N sentinel

#define SCAN_BS 256
#define SCAN_IT 16
#define SCAN_TILE (SCAN_BS * SCAN_IT)         // 4096
#define NB_EDGE ((EE + SCAN_TILE - 1) / SCAN_TILE)   // 1954
#define RDIG 16

#if __has_builtin(__builtin_amdgcn_tensor_load_to_lds)
#define HAVE_TDM 1
#else
#define HAVE_TDM 0
#endif
#if __has_include(<hip/amd_detail/amd_gfx1250_TDM.h>)
#define TDM_6ARG 1
#else
#define TDM_6ARG 0
#endif

typedef unsigned int v4ui __attribute__((ext_vector_type(4)));
typedef int v8si __attribute__((ext_vector_type(8)));
typedef int v4si __attribute__((ext_vector_type(4)));

// ---------------------------------------------------------------- CDNA5 async/TDM helpers
__device__ __forceinline__ void async_b128(u32 lds_off, u64 gaddr) {
  asm volatile("global_load_async_to_lds_b128 %0, %1, off"
               :: "v"(lds_off), "v"(gaddr) : "memory");
}
__device__ __forceinline__ void wait_async0() {
  asm volatile("s_wait_asynccnt 0" ::: "memory");
}
__device__ __forceinline__ void wait_tensor0() {
#if __has_builtin(__builtin_amdgcn_s_wait_tensorcnt)
  __builtin_amdgcn_s_wait_tensorcnt(0);
#else
  asm volatile("s_wait_tensorcnt 0" ::: "memory");
#endif
}

// ---------------------------------------------------------------- fills
__global__ void k_fill_u32(u32* p, long long n, u32 v) {
  long long i = (long long)blockIdx.x * 256 + threadIdx.x;
  if (i < n) p[i] = v;
}
__global__ void k_fill_i64(long long* p, long long n, long long v) {
  long long i = (long long)blockIdx.x * 256 + threadIdx.x;
  if (i < n) p[i] = v;
}

// ---------------------------------------------------------------- voxel keys
__global__ void k_voxkey(const float* __restrict__ pos, const int* __restrict__ batch,
                         u32* __restrict__ key_n, u32* __restrict__ P, int n) {
  int i = blockIdx.x * 256 + threadIdx.x;
  if (i >= n) return;
  float px = pos[3 * i + 0], py = pos[3 * i + 1], pt = pos[3 * i + 2];
  int qx = (int)floorf(px * (1.0f / 4.0f));
  int qy = (int)floorf(py * (1.0f / 4.0f));
  int qt = (int)floorf(pt * (1.0f / 4000.0f));
  int b = batch[i];
  u32 key = (((u32)(b * KXC + qx) * KYC + qy) * KTC + qt);
  key_n[i] = key;
  P[key] = 1u;
}

// ---------------------------------------------------------------- generic exclusive scan
__global__ void k_scan_bsum(const u32* __restrict__ in, long long n, u32* __restrict__ bsum) {
  __shared__ u32 red[SCAN_BS];
  int t = threadIdx.x;
  long long base = (long long)blockIdx.x * SCAN_TILE + t;
  u32 s = 0;
  for (int j = 0; j < SCAN_IT; ++j) {
    long long i = base + (long long)j * SCAN_BS;
    if (i < n) s += in[i];
  }
  red[t] = s;
  __syncthreads();
  for (int o = SCAN_BS / 2; o > 0; o >>= 1) {
    if (t < o) red[t] += red[t + o];
    __syncthreads();
  }
  if (t == 0) bsum[blockIdx.x] = red[0];
}

__global__ void k_scan_top(u32* data, int n, u32* total) {
  __shared__ u32 lds[1024];
  __shared__ u32 carry_s;
  int t = threadIdx.x;
  if (t == 0) carry_s = 0;
  __syncthreads();
  for (int base = 0; base < n; base += 1024) {
    int i = base + t;
    u32 v = (i < n) ? data[i] : 0u;
    lds[t] = v;
    __syncthreads();
    for (int o = 1; o < 1024; o <<= 1) {
      u32 a = (t >= o) ? lds[t - o] : 0u;
      __syncthreads();
      lds[t] += a;
      __syncthreads();
    }
    u32 excl = lds[t] - v;
    u32 c = carry_s;
    if (i < n) data[i] = c + excl;
    __syncthreads();
    if (t == 0) {
      int last = (n - base >= 1024) ? 1023 : (n - base - 1);
      carry_s = c + lds[last];
    }
    __syncthreads();
  }
  if (t == 0) *total = carry_s;
}

// exclusive scan apply; safe when in == out (each index read/written by same thread)
__global__ void k_scan_apply(const u32* __restrict__ in, u32* __restrict__ out,
                             long long n, const u32* __restrict__ bsum) {
  __shared__ u32 th[SCAN_BS];
  int t = threadIdx.x;
  long long base = (long long)blockIdx.x * SCAN_TILE + (long long)t * SCAN_IT;
  u32 v[SCAN_IT];
  u32 s = 0;
#pragma unroll
  for (int j = 0; j < SCAN_IT; ++j) {
    long long i = base + j;
    v[j] = (i < n) ? in[i] : 0u;
    s += v[j];
  }
  th[t] = s;
  __syncthreads();
  for (int o = 1; o < SCAN_BS; o <<= 1) {
    u32 a = (t >= o) ? th[t - o] : 0u;
    __syncthreads();
    th[t] += a;
    __syncthreads();
  }
  u32 run = bsum[blockIdx.x] + (th[t] - s);
#pragma unroll
  for (int j = 0; j < SCAN_IT; ++j) {
    long long i = base + j;
    if (i < n) out[i] = run;
    run += v[j];
  }
}

// ---------------------------------------------------------------- unique decode + inverse
__global__ void k_udecode(const u32* __restrict__ P, const u32* __restrict__ R,
                          long long* __restrict__ uniq, long long* __restrict__ nbatch) {
  int i = blockIdx.x * 256 + threadIdx.x;
  if (i >= KEYSPACE || !P[i]) return;
  u32 r = R[i];
  int qt = i & 63;
  int qy = (i >> 6) & 127;
  int qx = (i >> 13) & 127;
  int b  = i >> 20;
  uniq[(long long)r * 3 + 0] = qx;
  uniq[(long long)r * 3 + 1] = qy;
  uniq[(long long)r * 3 + 2] = qt;
  nbatch[r] = b;
}

__global__ void k_inv(const u32* __restrict__ key_n, const u32* __restrict__ R,
                      int* __restrict__ inv, int n) {
  int i = blockIdx.x * 256 + threadIdx.x;
  if (i < n) inv[i] = (int)R[key_n[i]];
}

// ---------------------------------------------------------------- pooling
// 64 nodes x 64 feats per block = 16KB LDS tile, DMA'd by the TDM (wave 0
// issues one tensor_load_to_lds; EXEC-independent per-wave op) or, without
// the builtin, per-lane async loads. Thread t consumes 16 floats.
__global__ void __launch_bounds__(256) k_pool(const float* __restrict__ x,
                                              const int* __restrict__ inv,
                                              float* __restrict__ pooled) {
  __shared__ float tile[64 * FF];
  int t = threadIdx.x;
  long long nodebase = (long long)blockIdx.x * 64;
  const float* gsrc = x + nodebase * FF;
#if HAVE_TDM
  if (t < 32) {  // one wave programs the DMA
    u32 lds_addr = (u32)(size_t)&tile[0];
    u64 ga = (u64)(size_t)gsrc;
    // D# group0: count=1 | lds_addr | global_addr[56:0] | type=2
    v4ui g0 = { 1u, lds_addr, (u32)ga,
                ((u32)(ga >> 32) & 0x01FFFFFFu) | 0x80000000u };
    // D# group1: data_size=4B; tensor_dim0=4096, tensor_dim1=1;
    //            tile_dim0=4096, tile_dim1=1; strides=4096
    v8si g1 = { 0x20000, 0x10000000, 0x00010000, 0x10000000,
                1, 4096, 0x10000000, 0 };
    v4si g2 = {0, 0, 0, 0};
    v4si g3 = {0, 0, 0, 0};
#if TDM_6ARG
    v8si g4 = {0, 0, 0, 0, 0, 0, 0, 0};
    __builtin_amdgcn_tensor_load_to_lds(g0, g1, g2, g3, g4, 0);
#else
    __builtin_amdgcn_tensor_load_to_lds(g0, g1, g2, g3, 0);
#endif
    wait_tensor0();
  }
  __syncthreads();
#else
  u32 ldsoff = (u32)(size_t)&tile[t * 16];
  u64 g = (u64)(size_t)(gsrc + (long long)t * 16);
#pragma unroll
  for (int k = 0; k < 4; ++k) async_b128(ldsoff + 16u * k, g + 16u * k);
  wait_async0();
#endif
  if (blockIdx.x + 1 < gridDim.x)
    __builtin_prefetch(x + (nodebase + 64) * FF + (long long)t * 16, 0, 1);
  int node = (int)nodebase + (t >> 2);
  int row = inv[node];
  int fbase = (t & 3) * 16;
  float* dst = pooled + (long long)row * FF + fbase;
#pragma unroll
  for (int j = 0; j < 16; ++j) {
    float v = tile[t * 16 + j];
    int iv = __float_as_int(v);
    if (iv >= 0) atomicMax((int*)&dst[j], iv);
    else         atomicMin((u32*)&dst[j], (u32)iv);
  }
}

__global__ void k_poolfix(float* pooled, const u32* Uptr, long long n) {
  long long i = (long long)blockIdx.x * 256 + threadIdx.x;
  if (i >= n) return;
  u32 U = *Uptr;
  if ((u32)(i >> 6) >= U) pooled[i] = 0.0f;
}

// ---------------------------------------------------------------- edges
__global__ void k_edgekey(const int* __restrict__ ei, const int* __restrict__ inv,
                          u64* __restrict__ ekey, int e) {
  int i = blockIdx.x * 256 + threadIdx.x;
  if (i >= e) return;
  int a = inv[ei[2 * i + 0]];
  int b = inv[ei[2 * i + 1]];
  ekey[i] = (a != b) ? ((u64)a * 1000000ull + (u64)b) : BIGKEY;
}

// radix histogram: stage 16 keys/thread into LDS via async b128,
// count 4-bit digits into per-wave privatized LDS counters
__global__ void __launch_bounds__(256) k_rhist(const u64* __restrict__ keys, int n, int shift,
                                               u32* __restrict__ hist, int nb) {
  __shared__ u64 tile[SCAN_TILE];
  __shared__ u32 cnt[8][RDIG];
  int t = threadIdx.x;
  int w = t >> 5;
  if ((t & 31) < RDIG) cnt[w][t & 31] = 0;
  __syncthreads();
  long long base = (long long)blockIdx.x * SCAN_TILE + (long long)t * SCAN_IT;
  if (base + SCAN_IT <= n) {
    u32 ldsoff = (u32)(size_t)&tile[t * SCAN_IT];
    u64 g = (u64)(size_t)&keys[base];
#pragma unroll
    for (int k = 0; k < 8; ++k) async_b128(ldsoff + 16u * k, g + 16u * k);
    wait_async0();
#pragma unroll
    for (int j = 0; j < SCAN_IT; ++j) {
      u32 d = (u32)((tile[t * SCAN_IT + j] >> shift) & 15ull);
      atomicAdd(&cnt[w][d], 1u);
    }
  } else {
    for (int j = 0; j < SCAN_IT; ++j) {
      long long i = base + j;
      if (i < n) {
        u32 d = (u32)((keys[i] >> shift) & 15ull);
        atomicAdd(&cnt[w][d], 1u);
      }
    }
  }
  __syncthreads();
  if (t < RDIG) {
    u32 s = 0;
#pragma unroll
    for (int ww = 0; ww < 8; ++ww) s += cnt[ww][t];
    hist[(long long)t * nb + blockIdx.x] = s;
  }
}

// stable scatter: per-(digit,thread) counters fully in LDS (single-owner
// columns, no atomics, no scratch arrays); scanned histogram base folded in.
__global__ void __launch_bounds__(256) k_rscatter(const u64* __restrict__ in, u64* __restrict__ out,
                                                  int n, int shift,
                                                  const u32* __restrict__ hist, int nb) {
  __shared__ u32 lds[RDIG * SCAN_BS];
  __shared__ u32 hb[RDIG];
  int t = threadIdx.x;
#pragma unroll
  for (int d = 0; d < RDIG; ++d) lds[d * SCAN_BS + t] = 0;
  if (t < RDIG) hb[t] = hist[(long long)t * nb + blockIdx.x];
  long long base = (long long)blockIdx.x * SCAN_TILE + (long long)t * SCAN_IT;
  u64 k[SCAN_IT];
  int m = (base + SCAN_IT <= n) ? SCAN_IT : (base < n ? (int)(n - base) : 0);
#pragma unroll
  for (int j = 0; j < SCAN_IT; ++j)
    if (j < m) {
      k[j] = in[base + j];
      lds[(u32)((k[j] >> shift) & 15ull) * SCAN_BS + t]++;
    }
  __syncthreads();
  if (t < RDIG) {  // serial exclusive scan over threads for digit t, + global base
    u32 run = hb[t];
    for (int b = 0; b < SCAN_BS; ++b) {
      u32 v = lds[t * SCAN_BS + b];
      lds[t * SCAN_BS + b] = run;
      run += v;
    }
  }
  __syncthreads();
#pragma unroll
  for (int j = 0; j < SCAN_IT; ++j)
    if (j < m) {
      u32 d = (u32)((k[j] >> shift) & 15ull);
      u32 p = lds[d * SCAN_BS + t]++;
      out[p] = k[j];
    }
}

__global__ void k_flags(const u64* __restrict__ keys, u32* __restrict__ flags, int n) {
  int i = blockIdx.x * 256 + threadIdx.x;
  if (i >= n) return;
  u64 kk = keys[i];
  flags[i] = (kk < BIGKEY && (i == 0 || keys[i - 1] != kk)) ? 1u : 0u;
}

__global__ void k_escatter(const u64* __restrict__ keys, const u32* __restrict__ flags,
                           const u32* __restrict__ pos, long long* __restrict__ eiout, int n) {
  int i = blockIdx.x * 256 + threadIdx.x;
  if (i >= n || !flags[i]) return;
  u64 kk = keys[i];
  u32 r = pos[i];
  eiout[(long long)r * 2 + 0] = (long long)(kk / 1000000ull);
  eiout[(long long)r * 2 + 1] = (long long)(kk % 1000000ull);
}

__global__ void k_selfloop(long long* __restrict__ eiout, const u32* Uptr, int n) {
  int i = blockIdx.x * 256 + threadIdx.x;
  if (i >= n) return;
  u32 U = *Uptr;
  long long v = ((u32)i < U) ? (long long)i : -1ll;
  eiout[((long long)EE + i) * 2 + 0] = v;
  eiout[((long long)EE + i) * 2 + 1] = v;
}

// ---------------------------------------------------------------- host side
static inline int gb(long long n) { return (int)((n + 255) / 256); }

static void scan_ex(const u32* in, u32* out, long long n, u32* bsum, u32* total,
                    hipStream_t s) {
  int nb = (int)((n + SCAN_TILE - 1) / SCAN_TILE);
  k_scan_bsum<<<nb, SCAN_BS, 0, s>>>(in, n, bsum);
  k_scan_top<<<1, 1024, 0, s>>>(bsum, nb, total);
  k_scan_apply<<<nb, SCAN_BS, 0, s>>>(in, out, n, bsum);
}

extern "C" void kernel_launch(void* const* d_in, const int* in_sizes, int n_in,
                              void* d_out, int out_size, void* d_ws, size_t ws_size,
                              hipStream_t stream) {
  (void)in_sizes; (void)n_in; (void)out_size; (void)ws_size;
  const float* x    = (const float*)d_in[0];
  const float* pos  = (const float*)d_in[1];
  const int*   eidx = (const int*)d_in[2];
  const int*   batch = (const int*)d_in[3];

  char* out = (char*)d_out;
  float*     pooled = (float*)out;
  long long* uniq   = (long long*)(out + (size_t)NN * FF * 4);
  long long* eiout  = uniq + (size_t)NN * 3;
  long long* nbatch = eiout + (size_t)(EE + NN) * 2;

  // workspace layout
  char* ws = (char*)d_ws;
  constexpr size_t SZ_P    = (size_t)KEYSPACE * 4;
  constexpr size_t OFF_P   = 0;
  constexpr size_t OFF_R   = OFF_P + SZ_P;
  constexpr size_t OFF_KEY = OFF_R + SZ_P;
  constexpr size_t OFF_INV = OFF_KEY + (size_t)NN * 4;
  constexpr size_t OFF_U   = OFF_INV + (size_t)NN * 4;
  constexpr size_t OFF_TD  = OFF_U + 256;
  constexpr size_t OFF_BS  = OFF_TD + 256;
  constexpr size_t OFF_H   = OFF_BS + (size_t)4096 * 4;
  constexpr size_t OFF_EKA = OFF_H + (((size_t)RDIG * NB_EDGE * 4 + 255) & ~(size_t)255);
  constexpr size_t OFF_EKB = OFF_EKA + (size_t)EE * 8;

  u32* P     = (u32*)(ws + OFF_P);
  u32* R     = (u32*)(ws + OFF_R);
  u32* key_n = (u32*)(ws + OFF_KEY);
  int* inv   = (int*)(ws + OFF_INV);
  u32* Uptr  = (u32*)(ws + OFF_U);
  u32* tdum  = (u32*)(ws + OFF_TD);
  u32* bsum  = (u32*)(ws + OFF_BS);
  u32* hist  = (u32*)(ws + OFF_H);
  u64* ekA   = (u64*)(ws + OFF_EKA);
  u64* ekB   = (u64*)(ws + OFF_EKB);
  u32* flags = (u32*)(ws + OFF_P);   // P dead by edge phase
  u32* epos  = (u32*)(ws + OFF_R);   // R dead by edge phase

  // ---- init outputs / presence
  k_fill_u32<<<gb(KEYSPACE), 256, 0, stream>>>(P, KEYSPACE, 0u);
  k_fill_u32<<<gb((long long)NN * FF), 256, 0, stream>>>((u32*)pooled, (long long)NN * FF,
                                                         0xFF800000u);   // -inf bits
  k_fill_i64<<<gb((long long)NN * 3), 256, 0, stream>>>(uniq, (long long)NN * 3, -1ll);
  k_fill_i64<<<gb((long long)(EE + NN) * 2), 256, 0, stream>>>(eiout, (long long)(EE + NN) * 2, -1ll);
  k_fill_i64<<<gb(NN), 256, 0, stream>>>(nbatch, NN, -1ll);

  // ---- voxel unique via dense bitmap + scan (L2-resident)
  k_voxkey<<<gb(NN), 256, 0, stream>>>(pos, batch, key_n, P, NN);
  scan_ex(P, R, KEYSPACE, bsum, Uptr, stream);
  k_udecode<<<KEYSPACE / 256, 256, 0, stream>>>(P, R, uniq, nbatch);
  k_inv<<<gb(NN), 256, 0, stream>>>(key_n, R, inv, NN);

  // ---- feature pooling (TDM/async LDS staging + atomic max)
  k_pool<<<NN / 64, 256, 0, stream>>>(x, inv, pooled);
  k_poolfix<<<gb((long long)NN * FF), 256, 0, stream>>>(pooled, Uptr, (long long)NN * FF);

  // ---- coarse edges: key, 40-bit LSD radix sort, dedup, scatter, self loops
  k_edgekey<<<gb(EE), 256, 0, stream>>>(eidx, inv, ekA, EE);
  u64* cur = ekA;
  u64* oth = ekB;
  for (int p = 0; p < 10; ++p) {
    int shift = 4 * p;
    k_rhist<<<NB_EDGE, 256, 0, stream>>>(cur, EE, shift, hist, NB_EDGE);
    scan_ex(hist, hist, (long long)RDIG * NB_EDGE, bsum, tdum, stream);
    k_rscatter<<<NB_EDGE, 256, 0, stream>>>(cur, oth, EE, shift, hist, NB_EDGE);
    u64* t = cur; cur = oth; oth = t;
  }
  // 10 passes (even) -> sorted result back in ekA (== cur)
  k_flags<<<gb(EE), 256, 0, stream>>>(cur, flags, EE);
  scan_ex(flags, epos, EE, bsum, tdum, stream);
  k_escatter<<<gb(EE), 256, 0, stream>>>(cur, flags, epos, eiout, EE);
  k_selfloop<<<gb(NN), 256, 0, stream>>>(eiout, Uptr, NN);
}